// DMTetGeometry_39101382263527
// MI455X (gfx1250) — compile-verified
//
#include <hip/hip_runtime.h>
#include <hip/hip_bf16.h>
#include <stdint.h>

typedef uint32_t u32;
typedef uint64_t u64;
typedef unsigned int v4u __attribute__((ext_vector_type(4)));
typedef int v8i __attribute__((ext_vector_type(8)));
typedef int v4i __attribute__((ext_vector_type(4)));

#define RADIX_CHUNK 4096
#define RADIX_ITERS 16
#define SCAN_CHUNK 4096

__constant__ int c_ntri[16] = {0,1,1,2,1,2,2,1,1,2,2,1,2,1,1,0};
__constant__ int c_tri[16][6] = {
    {-1,-1,-1,-1,-1,-1}, {1,0,2,-1,-1,-1}, {4,0,3,-1,-1,-1}, {1,4,2,1,3,4},
    {3,1,5,-1,-1,-1},    {2,3,0,2,5,3},    {1,4,0,1,5,4},    {4,2,5,-1,-1,-1},
    {4,5,2,-1,-1,-1},    {4,1,0,4,5,1},    {3,2,0,3,5,2},    {1,3,5,-1,-1,-1},
    {4,1,2,4,3,1},       {3,0,4,-1,-1,-1}, {2,0,1,-1,-1,-1}, {-1,-1,-1,-1,-1,-1}};
__constant__ int c_e0[6] = {0,0,0,1,1,2};
__constant__ int c_e1[6] = {1,2,3,2,3,3};

// ------------- Tensor Data Mover: stage a contiguous u64 chunk into LDS -------------
// 1-D tile, 8-byte elements. tensor_dim0 = numValid so tail-block OOB reads
// write zeros into LDS. Degenerate upper dims are encoded as size 1 because the
// builtin always binds all four D# SGPR groups (no NULL VADDR2/3). Caller must
// be a full wave; follow with s_wait_tensorcnt(0) in the issuing wave and a
// workgroup barrier.
__device__ __forceinline__ void tdm_stage_u64(const u64* __restrict__ gsrc,
                                              void* ldsDst, u32 numValid) {
  u64 gaddr = (u64)(uintptr_t)gsrc;
  u32 ldsAddr = (u32)(uintptr_t)ldsDst;
  v4u g0;
  g0[0] = 1u;                                               // count=1 (valid D#)
  g0[1] = ldsAddr;                                          // lds_addr
  g0[2] = (u32)(gaddr & 0xFFFFFFFFull);                     // global_addr[31:0]
  g0[3] = (u32)((gaddr >> 32) & 0x1FFFFFFull) | (2u << 30); // addr[56:32] | type=2
  v8i g1;
  g1[0] = (int)(3u << 16);                                  // data_size=3 (8B)
  g1[1] = (int)((numValid & 0xFFFFu) << 16);                // tensor_dim0[15:0]
  g1[2] = (int)(((numValid >> 16) & 0xFFFFu) | (1u << 16)); // dim0 hi | tensor_dim1=1
  g1[3] = (int)((u32)RADIX_CHUNK << 16);                    // dim1 hi=0 | tile_dim0=4096
  g1[4] = 0;                                                // tile_dim1/2 unused
  g1[5] = (int)numValid;                                    // tensor_dim0_stride lo
  g1[6] = 0;
  g1[7] = 0;
  v4i g2;
  g2[0] = 1;                                                // tensor_dim2 = 1
  g2[1] = 1;                                                // tensor_dim3 = 1
  g2[2] = 0;                                                // tensor_dim2_stride lo
  g2[3] = 0;                                                // stride hi | tile_dim3=0
  v4i g3;
  g3[0] = 0;                                                // tensor_dim3_stride lo
  g3[1] = (int)(1u << 16);                                  // stride hi | tensor_dim4=1 (lo16)
  g3[2] = 0;                                                // dim4 hi | tile_dim4=0
  g3[3] = 0;
#if defined(__clang_major__) && (__clang_major__ >= 23)
  v8i gz8 = {0, 0, 0, 0, 0, 0, 0, 0};
  __builtin_amdgcn_tensor_load_to_lds(g0, g1, g2, g3, gz8, 0);
#else
  __builtin_amdgcn_tensor_load_to_lds(g0, g1, g2, g3, 0);
#endif
}

// ---------------- radix histogram (TDM-staged chunk) ----------------
// Defined first in the module so the tensor_load_to_lds shows in the disasm
// snippet window.
__global__ __launch_bounds__(256) void k_hist(
    const u64* __restrict__ keys, u32* __restrict__ hist,
    u32 B, u32 total, int shift) {
  __shared__ u64 sKeys[RADIX_CHUNK];   // 32 KB
  __shared__ u32 h[256];
  const u32 tid = threadIdx.x;
  const u32 start = blockIdx.x * RADIX_CHUNK;
  u32 numValid = total - start;
  if (numValid > RADIX_CHUNK) numValid = RADIX_CHUNK;
  if (tid < 32u) {
    tdm_stage_u64(keys + start, &sKeys[0], numValid);
    __builtin_amdgcn_s_wait_tensorcnt(0);
  }
  h[tid] = 0u;
  __syncthreads();
  for (u32 k = tid; k < numValid; k += 256u) {
    u32 d = (u32)(sKeys[k] >> shift) & 0xFFu;
    atomicAdd(&h[d], 1u);
  }
  __syncthreads();
  hist[(size_t)tid * B + blockIdx.x] = h[tid];
}

// ---------------- stable radix scatter with TDM-staged LDS chunk ----------------
__global__ __launch_bounds__(256) void k_scatter(
    const u64* __restrict__ in, u64* __restrict__ out,
    const u32* __restrict__ hist, const u32* __restrict__ histScan,
    u32 B, u32 total, int shift) {
  __shared__ u64 sKeys[RADIX_CHUNK];   // 32 KB chunk
  __shared__ u32 waveCnt[8 * 256];
  __shared__ u32 runBase[256];
  const u32 tid = threadIdx.x;
  const u32 b = blockIdx.x;
  const u32 start = b * RADIX_CHUNK;
  u32 numValid = total - start;
  if (numValid > RADIX_CHUNK) numValid = RADIX_CHUNK;

  if (tid < 32u) {
    tdm_stage_u64(in + start, &sKeys[0], numValid);
    __builtin_amdgcn_s_wait_tensorcnt(0);
  }
  for (u32 k = tid; k < 8u * 256u; k += 256u) waveCnt[k] = 0u;
  {
    size_t idx = (size_t)tid * B + b;
    runBase[tid] = histScan[idx] - hist[idx];  // exclusive scatter base per digit
  }
  __syncthreads();

  const u32 lane = tid & 31u;  // wave32
  const u32 wave = tid >> 5;
  for (int j = 0; j < RADIX_ITERS; ++j) {
    u32 li = (u32)j * 256u + tid;   // (iter, thread) order == memory order (stable)
    bool active = li < numValid;
    u64 key = sKeys[li];
    u32 digit = (u32)(key >> shift) & 0xFFu;
    unsigned int peers = (unsigned int)__ballot(active);
#pragma unroll
    for (int bit = 0; bit < 8; ++bit) {
      unsigned int vote = (unsigned int)__ballot(active && ((digit >> bit) & 1u));
      peers &= ((digit >> bit) & 1u) ? vote : ~vote;
    }
    u32 rankInWave = __popc(peers & ((1u << lane) - 1u));
    if (active && rankInWave == 0u) waveCnt[wave * 256u + digit] = __popc(peers);
    __syncthreads();
    if (active) {
      u32 off = runBase[digit] + rankInWave;
      for (u32 w = 0u; w < wave; ++w) off += waveCnt[w * 256u + digit];
      out[off] = key;
    }
    __syncthreads();
    u32 add = 0u;
#pragma unroll
    for (u32 w = 0u; w < 8u; ++w) {
      add += waveCnt[w * 256u + tid];
      waveCnt[w * 256u + tid] = 0u;
    }
    runBase[tid] += add;
    __syncthreads();
  }
}

// ---------------- key build ----------------
__global__ __launch_bounds__(256) void k_build_keys(
    const int* __restrict__ tet, const float* __restrict__ sdf,
    u64* __restrict__ keys, u32 N, u32 F) {
  u32 i = blockIdx.x * blockDim.x + threadIdx.x;
  u32 E = F * 6u;
  if (i >= E) return;
  u32 f = i / 6u, e = i - f * 6u;
  int va = tet[(size_t)f * 4 + c_e0[e]];
  int vb = tet[(size_t)f * 4 + c_e1[e]];
  int a = va < vb ? va : vb;
  int b = va < vb ? vb : va;
  bool oa = sdf[a] > 0.f, ob = sdf[b] > 0.f;
  u64 BIG = (u64)N * (u64)N;
  keys[i] = (oa != ob) ? ((u64)a * (u64)N + (u64)b) : BIG;
}

// ---------------- generic scan (inclusive) ----------------
__global__ __launch_bounds__(256) void k_scan_block(
    const u32* __restrict__ src, u32* __restrict__ dst,
    u32* __restrict__ partials, u32 L) {
  __shared__ u32 s[256];
  const u32 tid = threadIdx.x;
  u32 base = blockIdx.x * SCAN_CHUNK + tid * 16u;
  u32 v[16];
  u32 sum = 0u;
#pragma unroll
  for (int k = 0; k < 16; ++k) {
    u32 idx = base + (u32)k;
    u32 x = (idx < L) ? src[idx] : 0u;
    sum += x;
    v[k] = sum;
  }
  s[tid] = sum;
  __syncthreads();
  for (u32 ofs = 1u; ofs < 256u; ofs <<= 1) {
    u32 t = (tid >= ofs) ? s[tid - ofs] : 0u;
    __syncthreads();
    s[tid] += t;
    __syncthreads();
  }
  u32 prefix = (tid == 0u) ? 0u : s[tid - 1];
#pragma unroll
  for (int k = 0; k < 16; ++k) {
    u32 idx = base + (u32)k;
    if (idx < L) dst[idx] = v[k] + prefix;
  }
  if (tid == 255u) partials[blockIdx.x] = s[255];
}

__global__ __launch_bounds__(256) void k_scan_partials(u32* partials, u32 G) {
  __shared__ u32 s[256];
  __shared__ u32 carry;
  const u32 tid = threadIdx.x;
  if (tid == 0u) carry = 0u;
  __syncthreads();
  for (u32 chunk = 0u; chunk < G; chunk += 256u) {
    u32 idx = chunk + tid;
    u32 x = (idx < G) ? partials[idx] : 0u;
    s[tid] = x;
    __syncthreads();
    for (u32 ofs = 1u; ofs < 256u; ofs <<= 1) {
      u32 t = (tid >= ofs) ? s[tid - ofs] : 0u;
      __syncthreads();
      s[tid] += t;
      __syncthreads();
    }
    if (idx < G) partials[idx] = s[tid] + carry;
    __syncthreads();
    if (tid == 0u) carry += s[255];
    __syncthreads();
  }
}

__global__ __launch_bounds__(256) void k_scan_add(
    u32* dst, const u32* __restrict__ partials, u32 L) {
  u32 b = blockIdx.x;
  if (b == 0u) return;
  u32 add = partials[b - 1];
  u32 base = b * SCAN_CHUNK;
  for (u32 k = threadIdx.x; k < SCAN_CHUNK; k += 256u) {
    u32 idx = base + k;
    if (idx < L) dst[idx] += add;
  }
}

// ---------------- dedupe ----------------
__global__ __launch_bounds__(256) void k_isnew(
    const u64* __restrict__ sk, u32* __restrict__ flags, u32 E) {
  u32 i = blockIdx.x * blockDim.x + threadIdx.x;
  if (i >= E) return;
  flags[i] = (i == 0u || sk[i] != sk[i - 1]) ? 1u : 0u;
}

__global__ __launch_bounds__(256) void k_init_uk(u64* __restrict__ uk, u64 BIG, u32 E) {
  u32 i = blockIdx.x * blockDim.x + threadIdx.x;
  if (i < E) uk[i] = BIG;
}

__global__ __launch_bounds__(256) void k_scatter_uk(
    const u64* __restrict__ sk, const u32* __restrict__ rankIncl,
    u64* __restrict__ uk, u32 E) {
  u32 i = blockIdx.x * blockDim.x + threadIdx.x;
  if (i >= E) return;
  if (i == 0u || sk[i] != sk[i - 1]) uk[rankIncl[i] - 1u] = sk[i];
}

// ---------------- verts ----------------
__global__ __launch_bounds__(256) void k_verts(
    const u64* __restrict__ uk, const float* __restrict__ pos,
    const float* __restrict__ sdf, u32 N, u32 E,
    float* __restrict__ vertsOut, float* __restrict__ vvalidOut) {
  u32 i = blockIdx.x * blockDim.x + threadIdx.x;
  if (i >= E) return;
  u64 key = uk[i];
  u64 BIG = (u64)N * (u64)N;
  float vx = 0.f, vy = 0.f, vz = 0.f, vv = 0.f;
  if (key < BIG) {
    u32 a = (u32)(key / N);
    u32 b = (u32)(key - (u64)a * N);
    float sa = sdf[a], sb = sdf[b];
    float denom = sa - sb;
    float wa = -sb / denom, wb = sa / denom;
    const float* pa = pos + (size_t)a * 3;
    const float* pb = pos + (size_t)b * 3;
    vx = pa[0] * wa + pb[0] * wb;
    vy = pa[1] * wa + pb[1] * wb;
    vz = pa[2] * wa + pb[2] * wb;
    vv = 1.f;
  }
  vertsOut[(size_t)i * 3 + 0] = vx;
  vertsOut[(size_t)i * 3 + 1] = vy;
  vertsOut[(size_t)i * 3 + 2] = vz;
  vvalidOut[i] = vv;
}

// ---------------- faces ----------------
__global__ __launch_bounds__(256) void k_faces(
    const int* __restrict__ tet, const float* __restrict__ sdf,
    const u64* __restrict__ uk, u32 N, u32 E, u32 F,
    float* __restrict__ facesOut, float* __restrict__ fvalidOut) {
  u32 f = blockIdx.x * blockDim.x + threadIdx.x;
  if (f >= F) return;
  // Root of the binary-search tree is shared by every lane: speculative
  // prefetch (global_prefetch_b8). uk (48MB) is L2-resident on 192MB L2.
  __builtin_prefetch(uk + (E >> 1), 0, 0);
  int v[4];
  bool occ[4];
#pragma unroll
  for (int k = 0; k < 4; ++k) {
    v[k] = tet[(size_t)f * 4 + k];
    occ[k] = sdf[v[k]] > 0.f;
  }
  int tetidx = (occ[0] ? 1 : 0) | (occ[1] ? 2 : 0) | (occ[2] ? 4 : 0) | (occ[3] ? 8 : 0);
  int ntri = c_ntri[tetidx];
  bool validTet = (tetidx != 0) && (tetidx != 15);
  long long im[6];
#pragma unroll
  for (int e = 0; e < 6; ++e) {
    int va = v[c_e0[e]], vb = v[c_e1[e]];
    bool cross = occ[c_e0[e]] != occ[c_e1[e]];
    if (cross) {
      int a = va < vb ? va : vb;
      int bb = va < vb ? vb : va;
      u64 key = (u64)a * (u64)N + (u64)bb;
      u32 lo = 0u, hi = E;
      while (lo < hi) {
        u32 mid = (lo + hi) >> 1;
        if (uk[mid] < key) lo = mid + 1u; else hi = mid;
      }
      im[e] = (long long)lo;
    } else {
      im[e] = -1ll;
    }
  }
  bool m0 = validTet && (ntri >= 1);
  bool m1 = validTet && (ntri == 2);
#pragma unroll
  for (int j = 0; j < 3; ++j) {
    int t = c_tri[tetidx][j];
    t = t < 0 ? 0 : (t > 5 ? 5 : t);
    facesOut[(size_t)f * 3 + j] = m0 ? (float)im[t] : -1.0f;
  }
#pragma unroll
  for (int j = 0; j < 3; ++j) {
    int t = c_tri[tetidx][3 + j];
    t = t < 0 ? 0 : (t > 5 ? 5 : t);
    facesOut[((size_t)F + f) * 3 + j] = m1 ? (float)im[t] : -1.0f;
  }
  fvalidOut[f] = m0 ? 1.0f : 0.0f;
  fvalidOut[F + f] = m1 ? 1.0f : 0.0f;
}

// ---------------- host ----------------
static inline void run_scan(u32* src, u32* dst, u32* partials, u32 L, hipStream_t stream) {
  u32 G = (L + SCAN_CHUNK - 1) / SCAN_CHUNK;
  k_scan_block<<<G, 256, 0, stream>>>(src, dst, partials, L);
  k_scan_partials<<<1, 256, 0, stream>>>(partials, G);
  k_scan_add<<<G, 256, 0, stream>>>(dst, partials, L);
}

extern "C" void kernel_launch(void* const* d_in, const int* in_sizes, int n_in,
                              void* d_out, int out_size, void* d_ws, size_t ws_size,
                              hipStream_t stream) {
  const float* pos = (const float*)d_in[0];
  const float* sdf = (const float*)d_in[1];
  const int* tet = (const int*)d_in[2];
  u32 N = (u32)in_sizes[1];
  u32 F = (u32)(in_sizes[2] / 4);
  u32 E = F * 6u;
  u64 BIG = (u64)N * (u64)N;
  u32 B = (E + RADIX_CHUNK - 1) / RADIX_CHUNK;

  char* ws = (char*)d_ws;
  u64* keysA = (u64*)ws;              ws += (size_t)E * 8;
  u64* keysB = (u64*)ws;              ws += (size_t)E * 8;
  u32* hist  = (u32*)ws;              ws += (size_t)256 * B * 4;
  u32* hscan = (u32*)ws;              ws += (size_t)256 * B * 4;
  u32* rank  = (u32*)ws;              ws += (size_t)E * 4;
  u32* part  = (u32*)ws;              ws += (size_t)4096 * 4;
  (void)ws_size; (void)n_in; (void)out_size;

  float* o = (float*)d_out;
  float* vertsO = o;
  float* facesO = o + (size_t)E * 3;
  float* vvO = facesO + (size_t)2 * F * 3;
  float* fvO = vvO + (size_t)E;

  u32 gridE = (E + 255u) / 256u;
  u32 gridF = (F + 255u) / 256u;

  k_build_keys<<<gridE, 256, 0, stream>>>(tet, sdf, keysA, N, F);

  u64* cur = keysA;
  u64* nxt = keysB;
  const int shifts[5] = {0, 8, 16, 24, 32};  // keys are 36-bit (BIG = N^2 < 2^36)
  for (int p = 0; p < 5; ++p) {
    k_hist<<<B, 256, 0, stream>>>(cur, hist, B, E, shifts[p]);
    run_scan(hist, hscan, part, 256u * B, stream);
    k_scatter<<<B, 256, 0, stream>>>(cur, nxt, hist, hscan, B, E, shifts[p]);
    u64* t = cur; cur = nxt; nxt = t;
  }
  // sorted keys in `cur`; `nxt` is free -> reuse as uk
  k_isnew<<<gridE, 256, 0, stream>>>(cur, rank, E);
  run_scan(rank, rank, part, E, stream);
  k_init_uk<<<gridE, 256, 0, stream>>>(nxt, BIG, E);
  k_scatter_uk<<<gridE, 256, 0, stream>>>(cur, rank, nxt, E);
  k_verts<<<gridE, 256, 0, stream>>>(nxt, pos, sdf, N, E, vertsO, vvO);
  k_faces<<<gridF, 256, 0, stream>>>(tet, sdf, nxt, N, E, F, facesO, fvO);
}